// BoundaryLoss_81784767250654
// MI455X (gfx1250) — compile-verified
//
#include <hip/hip_runtime.h>
#include <hip/hip_bf16.h>

#ifndef __has_builtin
#define __has_builtin(x) 0
#endif

// Problem constants (from setup_inputs): [B, C, D, H, W] = [4, 2, 128, 192, 192]
#define Bk 4
#define Ck 2
#define Dk 128
#define Hk 192
#define Wk 192
#define EPSk 1e-7f

// Tiling: 256 threads own a 32(W) x 8(H) tile, marching DCHUNK planes in D.
#define TW 32
#define TH 8
#define HLW 34          // tile width + halo
#define HLH 10          // tile height + halo
#define HALO_N (HLW * HLH)   // 340
#define DCHUNK 32
#define TILES_W (Wk / TW)    // 6
#define TILES_H (Hk / TH)    // 24
#define NPART (Bk * (Dk / DCHUNK) * TILES_W * TILES_H)  // 2304 partial sums

// ---- CDNA5 async global->LDS path (probe-verified on this toolchain) ----
#if defined(__gfx1250__) && __has_builtin(__builtin_amdgcn_global_load_async_to_lds_b32)
#define USE_ASYNC_LDS 1
#else
#define USE_ASYNC_LDS 0
#endif

__device__ __forceinline__ void async_ld_i32(const int* g, int* l) {
#if USE_ASYNC_LDS
  __builtin_amdgcn_global_load_async_to_lds_b32(
      (__attribute__((address_space(1))) int*)g,
      (__attribute__((address_space(3))) int*)l,
      /*offset=*/0, /*cpol=*/0);
#else
  *l = *g;
#endif
}

__device__ __forceinline__ void wait_async_lds() {
#if USE_ASYNC_LDS
#if __has_builtin(__builtin_amdgcn_s_wait_asynccnt)
  __builtin_amdgcn_s_wait_asynccnt(0);
#else
  asm volatile("s_wait_asynccnt 0x0" ::: "memory");
#endif
#endif
}

// Load one halo element (both channels) of one D-plane into LDS ring slot.
// sP gets sigmoid(x) (0.0f sentinel when out of volume -> mask 0, matching the
// zero-padded erosion). sT gets the raw int target (0 when OOB).
__device__ __forceinline__ void load_elem(
    int slot, int d, bool dok, int b, int h0, int w0, int e,
    const float* __restrict__ x, const int* __restrict__ t,
    float (*sP)[Ck][HLH][HLW], int (*sT)[Ck][HLH][HLW])
{
  const int hh = e / HLW;
  const int ww = e - hh * HLW;
  const int h = h0 - 1 + hh;
  const int w = w0 - 1 + ww;
  const bool ok = dok && ((unsigned)h < (unsigned)Hk) && ((unsigned)w < (unsigned)Wk);
#pragma unroll
  for (int c = 0; c < Ck; ++c) {
    if (ok) {
      const int idx = (((b * Ck + c) * Dk + d) * Hk + h) * Wk + w;
      const float xv = x[idx];
      sP[slot][c][hh][ww] = 1.0f / (1.0f + expf(-xv));   // sigmoid
      async_ld_i32(&t[idx], &sT[slot][c][hh][ww]);        // async DMA to LDS
    } else {
      sP[slot][c][hh][ww] = 0.0f;
      sT[slot][c][hh][ww] = 0;
    }
  }
}

__device__ __forceinline__ void load_plane(
    int slot, int d, int b, int h0, int w0,
    const float* __restrict__ x, const int* __restrict__ t,
    float (*sP)[Ck][HLH][HLW], int (*sT)[Ck][HLH][HLW], int tid)
{
  const bool dok = ((unsigned)d < (unsigned)Dk);
  load_elem(slot, d, dok, b, h0, w0, tid, x, t, sP, sT);
  if (tid < HALO_N - 256)   // 84 threads cover elements 256..339
    load_elem(slot, d, dok, b, h0, w0, tid + 256, x, t, sP, sT);
}

__global__ __launch_bounds__(256)
void BoundaryLoss_main_kernel(const float* __restrict__ x,
                              const int* __restrict__ t,
                              double* __restrict__ partials)
{
  // 4-slot ring: plane (d0+i) lives in slot (i+1)&3. With 4 slots the writer
  // of iteration i touches slot (i+2)&3 while readers (bounded to one barrier
  // interval of skew) touch only {i-1, i, i+1} mod 4 -> a single barrier per
  // plane is sufficient.
  __shared__ float sP[4][Ck][HLH][HLW];
  __shared__ int   sT[4][Ck][HLH][HLW];
  __shared__ float warr[8];

  const int tid = threadIdx.x;
  const int tileId = blockIdx.x;                 // 0 .. TILES_W*TILES_H-1
  const int tw = tileId % TILES_W;
  const int th = tileId / TILES_W;
  const int b  = blockIdx.z;
  const int w0 = tw * TW;
  const int h0 = th * TH;
  const int d0 = blockIdx.y * DCHUNK;

  // Prime the ring with planes d0-1 (slot 0) and d0 (slot 1).
  load_plane(0, d0 - 1, b, h0, w0, x, t, sP, sT, tid);
  load_plane(1, d0,     b, h0, w0, x, t, sP, sT, tid);

  const int tx = tid & (TW - 1);   // w within tile
  const int ty = tid >> 5;         // h within tile
  float acc = 0.0f;

  for (int i = 0; i < DCHUNK; ++i) {
    const int prv = i & 3;
    const int cur = (i + 1) & 3;
    const int nxt = (i + 2) & 3;

    // Stage next plane (d0+i+1); then one barrier makes it visible.
    load_plane(nxt, d0 + i + 1, b, h0, w0, x, t, sP, sT, tid);
    wait_async_lds();
    __syncthreads();

    // Process plane d0+i at voxel (ty, tx) of the tile.
    float bp = 0.0f, bt = 0.0f;
#pragma unroll
    for (int c = 0; c < Ck; ++c) {
      const float pc = sP[cur][c][ty + 1][tx + 1];
      const bool ep = (pc != 0.0f)
          && (sP[cur][c][ty + 1][tx    ] != 0.0f)
          && (sP[cur][c][ty + 1][tx + 2] != 0.0f)
          && (sP[cur][c][ty    ][tx + 1] != 0.0f)
          && (sP[cur][c][ty + 2][tx + 1] != 0.0f)
          && (sP[prv][c][ty + 1][tx + 1] != 0.0f)
          && (sP[nxt][c][ty + 1][tx + 1] != 0.0f);
      bp += pc - (ep ? 1.0f : 0.0f);

      const int tc = sT[cur][c][ty + 1][tx + 1];
      const bool et = (tc != 0)
          && (sT[cur][c][ty + 1][tx    ] != 0)
          && (sT[cur][c][ty + 1][tx + 2] != 0)
          && (sT[cur][c][ty    ][tx + 1] != 0)
          && (sT[cur][c][ty + 2][tx + 1] != 0)
          && (sT[prv][c][ty + 1][tx + 1] != 0)
          && (sT[nxt][c][ty + 1][tx + 1] != 0);
      bt += (float)tc - (et ? 1.0f : 0.0f);
    }

    const float bi = fminf(fmaxf(bp, EPSk), 1.0f - EPSk);
    acc += -(bt * logf(bi) + (1.0f - bt) * log1pf(-bi));
  }

  // Deterministic block reduction: wave32 shuffle tree, then 8 wave sums.
#pragma unroll
  for (int off = 16; off > 0; off >>= 1)
    acc += __shfl_down(acc, off, 32);
  if ((tid & 31) == 0) warr[tid >> 5] = acc;
  __syncthreads();
  if (tid == 0) {
    float s = 0.0f;
#pragma unroll
    for (int k = 0; k < 8; ++k) s += warr[k];
    const int pid = (blockIdx.z * gridDim.y + blockIdx.y) * gridDim.x + blockIdx.x;
    partials[pid] = (double)s;
  }
}

__global__ __launch_bounds__(256)
void BoundaryLoss_finalize_kernel(const double* __restrict__ partials,
                                  int n, float* __restrict__ out)
{
  __shared__ double sm[256];
  double s = 0.0;
  for (int i = threadIdx.x; i < n; i += 256) s += partials[i];  // fixed order
  sm[threadIdx.x] = s;
  __syncthreads();
  for (int st = 128; st > 0; st >>= 1) {
    if (threadIdx.x < st) sm[threadIdx.x] += sm[threadIdx.x + st];
    __syncthreads();
  }
  if (threadIdx.x == 0) {
    const double nvox = (double)Bk * (double)Dk * (double)Hk * (double)Wk;
    out[0] = (float)(sm[0] / nvox);
  }
}

extern "C" void kernel_launch(void* const* d_in, const int* in_sizes, int n_in,
                              void* d_out, int out_size, void* d_ws, size_t ws_size,
                              hipStream_t stream) {
  const float* x = (const float*)d_in[0];   // inputs  [4,2,128,192,192] f32
  const int*   t = (const int*)d_in[1];     // targets [4,2,128,192,192] i32
  double* partials = (double*)d_ws;         // NPART doubles (fully overwritten)
  float* out = (float*)d_out;

  dim3 grid(TILES_W * TILES_H, Dk / DCHUNK, Bk);  // (144, 4, 4) = 2304 blocks
  BoundaryLoss_main_kernel<<<grid, dim3(256), 0, stream>>>(x, t, partials);
  BoundaryLoss_finalize_kernel<<<1, 256, 0, stream>>>(partials, NPART, out);
}